// GraphSAGE_64845416235694
// MI455X (gfx1250) — compile-verified
//
#include <hip/hip_runtime.h>
#include <math.h>

typedef __attribute__((ext_vector_type(2))) float v2f;
typedef __attribute__((ext_vector_type(8))) float v8f;

#define N_NODES 50000
#define N_EDGES 300000
#define D_FEAT  256
#define N_TILES (N_NODES / 16)   // 3125

// ---------------- utility ----------------

__global__ void zero_kernel(float* __restrict__ p, long n) {
  long i = (long)blockIdx.x * blockDim.x + threadIdx.x;
  long stride = (long)gridDim.x * blockDim.x;
  for (; i < n; i += stride) p[i] = 0.0f;
}

__global__ void degree_kernel(const int* __restrict__ ei, float* __restrict__ cnt) {
  int e = blockIdx.x * blockDim.x + threadIdx.x;
  if (e < N_EDGES) atomicAdd(&cnt[ei[N_EDGES + e]], 1.0f);
}

// ---------------- GEMM 1: y3[N,64] = x[N,256] @ [Wl3 | Wr3] ----------------
// Weights staged in LDS (stride 72 => the two 16-lane halves hit disjoint
// bank ranges). Each wave computes 4 row-tiles x 64 cols: 16 WMMAs per
// K-step against 4 A-fragments + 4 B-fragments (4x B reuse).

__global__ void __launch_bounds__(256)
gemm1_kernel(const float* __restrict__ x,
             const float* __restrict__ Wl,
             const float* __restrict__ Wr,
             float* __restrict__ y3) {
  __shared__ float sW[256 * 72];   // 72 KB, [k][n] padded
  for (int idx = threadIdx.x; idx < 256 * 64; idx += blockDim.x) {
    int k = idx >> 6, n = idx & 63;
    sW[k * 72 + n] = (n < 32) ? Wl[k * 32 + n] : Wr[k * 32 + (n - 32)];
  }
  __syncthreads();

  int wtask = blockIdx.x * (blockDim.x >> 5) + (threadIdx.x >> 5);
  int tile0 = wtask * 4;
  if (tile0 >= N_TILES) return;
  int lane = threadIdx.x & 31;
  int half = lane >> 4;       // 0: K {0,1}, 1: K {2,3}
  int l15  = lane & 15;

  const float* arow[4];
#pragma unroll
  for (int i = 0; i < 4; ++i) {
    long r = (long)(tile0 + i) * 16 + l15;
    if (r > (long)N_NODES - 1) r = N_NODES - 1;   // clamp OOB tail tiles
    arow[i] = x + r * D_FEAT + half * 2;
  }

  v8f c[4][4] = {};

  for (int k = 0; k < D_FEAT; k += 4) {
    v2f a[4], b[4];
#pragma unroll
    for (int i = 0; i < 4; ++i) {
      __builtin_prefetch(arow[i] + k + 64, 0, 0);   // global_prefetch_b8
      a[i].x = arow[i][k];
      a[i].y = arow[i][k + 1];
    }
    const float* w0 = sW + (k + half * 2) * 72 + l15;
#pragma unroll
    for (int nt = 0; nt < 4; ++nt) {
      b[nt].x = w0[nt * 16];        // row k+half*2
      b[nt].y = w0[nt * 16 + 72];   // row k+half*2+1
    }
#pragma unroll
    for (int i = 0; i < 4; ++i)
#pragma unroll
      for (int nt = 0; nt < 4; ++nt)
        c[i][nt] = __builtin_amdgcn_wmma_f32_16x16x4_f32(
            false, a[i], false, b[nt], (short)0, c[i][nt], false, false);
  }

#pragma unroll
  for (int i = 0; i < 4; ++i) {
    if (tile0 + i >= N_TILES) break;
#pragma unroll
    for (int r = 0; r < 8; ++r) {
      int row = (tile0 + i) * 16 + r + half * 8;
      float* yr = y3 + (long)row * 64 + l15;
      yr[0]  = c[i][0][r];
      yr[16] = c[i][1][r];
      yr[32] = c[i][2][r];
      yr[48] = c[i][3][r];
    }
  }
}

// ---------------- GEMM 2: y4[N,80] = h3[N,32] @ [Wl4 | Wr4] ----------------
// Packs weights straight into LDS (10 KB); 2 row-tiles per wave.

__global__ void __launch_bounds__(256)
gemm2_kernel(const float* __restrict__ h3,
             const float* __restrict__ Wl4,
             const float* __restrict__ Wr4,
             float* __restrict__ y4) {
  __shared__ float sW[32 * 80];
  for (int idx = threadIdx.x; idx < 32 * 80; idx += blockDim.x) {
    int k = idx / 80, n = idx % 80;
    sW[idx] = (n < 40) ? Wl4[k * 40 + n] : Wr4[k * 40 + (n - 40)];
  }
  __syncthreads();

  int wtask = blockIdx.x * (blockDim.x >> 5) + (threadIdx.x >> 5);
  int tile0 = wtask * 2;
  if (tile0 >= N_TILES) return;
  int lane = threadIdx.x & 31;
  int half = lane >> 4;
  int l15  = lane & 15;

  const float* arow[2];
#pragma unroll
  for (int i = 0; i < 2; ++i) {
    long r = (long)(tile0 + i) * 16 + l15;
    if (r > (long)N_NODES - 1) r = N_NODES - 1;
    arow[i] = h3 + r * 32 + half * 2;
  }

  v8f c[2][5] = {};

#pragma unroll
  for (int k = 0; k < 32; k += 4) {
    v2f a[2], b[5];
#pragma unroll
    for (int i = 0; i < 2; ++i) {
      a[i].x = arow[i][k];
      a[i].y = arow[i][k + 1];
    }
    const float* w0 = sW + (k + half * 2) * 80 + l15;
#pragma unroll
    for (int nt = 0; nt < 5; ++nt) {
      b[nt].x = w0[nt * 16];
      b[nt].y = w0[nt * 16 + 80];
    }
#pragma unroll
    for (int i = 0; i < 2; ++i)
#pragma unroll
      for (int nt = 0; nt < 5; ++nt)
        c[i][nt] = __builtin_amdgcn_wmma_f32_16x16x4_f32(
            false, a[i], false, b[nt], (short)0, c[i][nt], false, false);
  }

#pragma unroll
  for (int i = 0; i < 2; ++i) {
    if (tile0 + i >= N_TILES) break;
#pragma unroll
    for (int r = 0; r < 8; ++r) {
      int row = (tile0 + i) * 16 + r + half * 8;
      float* yr = y4 + (long)row * 80 + l15;
#pragma unroll
      for (int nt = 0; nt < 5; ++nt) yr[nt * 16] = c[i][nt][r];
    }
  }
}

// ---------------- edge scatter (post-projection: 8x less traffic) ----------------

__global__ void scatter3_kernel(const int* __restrict__ ei,
                                const float* __restrict__ y3,
                                float* __restrict__ agg3) {
  long t = (long)blockIdx.x * blockDim.x + threadIdx.x;
  if (t >= (long)N_EDGES * 8) return;
  int e = (int)(t >> 3);
  int c = (int)(t & 7) * 4;
  int s = ei[e], d = ei[N_EDGES + e];
  const float4 v = *(const float4*)(y3 + (long)s * 64 + c);
  float* ap = agg3 + (long)d * 32 + c;
  atomicAdd(ap + 0, v.x);
  atomicAdd(ap + 1, v.y);
  atomicAdd(ap + 2, v.z);
  atomicAdd(ap + 3, v.w);
}

__global__ void scatter4_kernel(const int* __restrict__ ei,
                                const float* __restrict__ y4,
                                float* __restrict__ agg4) {
  long t = (long)blockIdx.x * blockDim.x + threadIdx.x;
  if (t >= (long)N_EDGES * 10) return;
  int e = (int)(t / 10);
  int c = (int)(t % 10) * 4;
  int s = ei[e], d = ei[N_EDGES + e];
  const float4 v = *(const float4*)(y4 + (long)s * 80 + c);
  float* ap = agg4 + (long)d * 40 + c;
  atomicAdd(ap + 0, v.x);
  atomicAdd(ap + 1, v.y);
  atomicAdd(ap + 2, v.z);
  atomicAdd(ap + 3, v.w);
}

// ---------------- finalize ----------------

__global__ void finalize3_kernel(const float* __restrict__ agg3,
                                 const float* __restrict__ y3,
                                 const float* __restrict__ b3,
                                 const float* __restrict__ cnt,
                                 float* __restrict__ h3) {
  long t = (long)blockIdx.x * blockDim.x + threadIdx.x;
  if (t >= (long)N_NODES * 32) return;
  int i = (int)(t >> 5);
  int j = (int)(t & 31);
  float inv = 1.0f / fmaxf(cnt[i], 1.0f);
  float v = agg3[t] * inv + y3[(long)i * 64 + 32 + j] + b3[j];
  h3[t] = fmaxf(v, 0.0f);
}

__global__ void finalize4_kernel(const float* __restrict__ agg4,
                                 const float* __restrict__ y4,
                                 const float* __restrict__ b4,
                                 const float* __restrict__ cnt,
                                 float* __restrict__ out) {
  int i = blockIdx.x * blockDim.x + threadIdx.x;
  if (i >= N_NODES) return;
  float inv = 1.0f / fmaxf(cnt[i], 1.0f);
  float h[40];
  float m = -1e30f;
#pragma unroll
  for (int j = 0; j < 40; ++j) {
    float v = agg4[(long)i * 40 + j] * inv + y4[(long)i * 80 + 40 + j] + b4[j];
    h[j] = v;
    m = fmaxf(m, v);
  }
  float s = 0.0f;
#pragma unroll
  for (int j = 0; j < 40; ++j) s += __expf(h[j] - m);
  float lse = m + __logf(s);
#pragma unroll
  for (int j = 0; j < 40; ++j) out[(long)i * 40 + j] = h[j] - lse;
}

// ---------------- launch ----------------

extern "C" void kernel_launch(void* const* d_in, const int* in_sizes, int n_in,
                              void* d_out, int out_size, void* d_ws, size_t ws_size,
                              hipStream_t stream) {
  const float* x   = (const float*)d_in[0];
  const int*   ei  = (const int*)d_in[1];
  const float* Wl3 = (const float*)d_in[8];
  const float* Wr3 = (const float*)d_in[9];
  const float* b3  = (const float*)d_in[10];
  const float* Wl4 = (const float*)d_in[11];
  const float* Wr4 = (const float*)d_in[12];
  const float* b4  = (const float*)d_in[13];
  float* out = (float*)d_out;

  const long N = N_NODES;
  float* ws   = (float*)d_ws;
  float* y3   = ws;               // 64N
  float* cnt  = y3 + 64 * N;      // N
  float* agg3 = cnt + N;          // 32N
  float* h3   = agg3 + 32 * N;    // 32N
  float* y4   = h3 + 32 * N;      // 80N
  float* agg4 = y4 + 80 * N;      // 40N

  const int BT = 256;
  // zero cnt + agg3 (contiguous 33N) and agg4 (40N)
  zero_kernel<<<1024, BT, 0, stream>>>(cnt, 33 * N);
  zero_kernel<<<1024, BT, 0, stream>>>(agg4, 40 * N);

  degree_kernel<<<(N_EDGES + BT - 1) / BT, BT, 0, stream>>>(ei, cnt);

  // GEMM1: 3125 tiles / 4 per wave = 782 wave-tasks, 8 waves per block
  int wt1 = (N_TILES + 3) / 4;
  gemm1_kernel<<<(wt1 + 7) / 8, BT, 0, stream>>>(x, Wl3, Wr3, y3);

  long s3 = (long)N_EDGES * 8;
  scatter3_kernel<<<(int)((s3 + BT - 1) / BT), BT, 0, stream>>>(ei, y3, agg3);

  long f3 = (long)N_NODES * 32;
  finalize3_kernel<<<(int)((f3 + BT - 1) / BT), BT, 0, stream>>>(agg3, y3, b3, cnt, h3);

  // GEMM2: 3125 tiles / 2 per wave = 1563 wave-tasks
  int wt2 = (N_TILES + 1) / 2;
  gemm2_kernel<<<(wt2 + 7) / 8, BT, 0, stream>>>(h3, Wl4, Wr4, y4);

  long s4 = (long)N_EDGES * 10;
  scatter4_kernel<<<(int)((s4 + BT - 1) / BT), BT, 0, stream>>>(ei, y4, agg4);

  finalize4_kernel<<<(N_NODES + BT - 1) / BT, BT, 0, stream>>>(agg4, y4, b4, cnt, out);
}